// Voxelization_15135464751874
// MI455X (gfx1250) — compile-verified
//
#include <hip/hip_runtime.h>
#include <cstdint>
#include <cstddef>

#define RES   32
#define R3    32768            // 32^3
#define BATCH 8
#define CH    64
#define GRID_ELEMS ((size_t)BATCH * CH * R3)   // 16,777,216

// ---- workspace layout (32-bit words) ----
#define WS_COUNTS 0                        // BATCH*R3 floats
#define WS_SUMS   (BATCH * R3)             // BATCH*4 floats (xyz sums)
#define WS_MAXN   (WS_SUMS + BATCH * 4)    // BATCH uints (max-norm bits)
#define WS_VREL   (WS_MAXN + 32)           // BATCH*N ints (voxel index per point)
#define WS_INIT_WORDS WS_VREL

typedef __attribute__((ext_vector_type(2))) float v2f;
typedef __attribute__((ext_vector_type(8))) float v8f;

// ---------- gfx1250 async LDS->global store (inline asm; CDNA5 ISA) ----------
__device__ __forceinline__ void async_store_lds_to_global_b128(void* gptr, const void* lptr) {
  unsigned long long ga = (unsigned long long)(uintptr_t)gptr;
  // flat shared-aperture address: low 32 bits are the LDS byte offset (ISA 10.2)
  unsigned la = (unsigned)(uintptr_t)lptr;
  asm volatile("global_store_async_from_lds_b128 %0, %1, off"
               :: "v"(ga), "v"(la) : "memory");
}
__device__ __forceinline__ void wait_asynccnt0() {
  asm volatile("s_wait_asynccnt 0" ::: "memory");
}

// ---------- kernel 0: zero the stat/count region of ws ----------
__global__ void k_init(float* __restrict__ ws) {
  int i = blockIdx.x * blockDim.x + threadIdx.x;
  if (i < WS_INIT_WORDS) ws[i] = 0.0f;
}

// ---------- kernel 1: per-batch coordinate sums via V_WMMA_F32_16X16X4_F32 ----------
// sum(x) == ones-matmul: D = A x B (+C) with B = all-ones accumulates 64 f32/wave/issue.
// Every column of D holds the 16 row-sums, so grand total = (sum of all D)/16 (exact).
__global__ void k_sums(const float* __restrict__ coords, float* __restrict__ ws, int N) {
  const int b = blockIdx.y;
  const int tid = threadIdx.x;
  const int lane = tid & 31;
  const int wave = tid >> 5;
  const int nchunk = (N + 63) >> 6;                       // 64 floats per WMMA
  const int wstride = gridDim.x * (blockDim.x >> 5);      // total waves in this row
  const int w0 = blockIdx.x * (blockDim.x >> 5) + wave;
  __shared__ float red[256];
  const v2f bones = {1.0f, 1.0f};                          // B = all ones (4x16)
  for (int a = 0; a < 3; ++a) {
    const float* p = coords + (size_t)(b * 3 + a) * N;
    v8f c = {};
    for (int ch = w0; ch < nchunk; ch += wstride) {
      int i0 = (ch << 6) + lane;
      int i1 = i0 + 32;
      v2f av;
      av.x = (i0 < N) ? p[i0] : 0.0f;   // predicated select keeps EXEC all-ones
      av.y = (i1 < N) ? p[i1] : 0.0f;
      c = __builtin_amdgcn_wmma_f32_16x16x4_f32(false, av, false, bones,
                                                (short)0, c, false, false);
    }
    float s = c[0] + c[1] + c[2] + c[3] + c[4] + c[5] + c[6] + c[7];
    red[tid] = s;
    __syncthreads();
    for (int off = 128; off > 0; off >>= 1) {
      if (tid < off) red[tid] += red[tid + off];
      __syncthreads();
    }
    if (tid == 0) atomicAdd(&ws[WS_SUMS + b * 4 + a], red[0] * 0.0625f);  // /16 exact
    __syncthreads();
  }
}

// ---------- kernel 2: per-batch max centered-norm (scale) ----------
__global__ void k_maxnorm(const float* __restrict__ coords, float* __restrict__ ws, int N) {
  const int b = blockIdx.y;
  const int tid = threadIdx.x;
  const float invN = 1.0f / (float)N;
  const float mx = ws[WS_SUMS + b * 4 + 0] * invN;
  const float my = ws[WS_SUMS + b * 4 + 1] * invN;
  const float mz = ws[WS_SUMS + b * 4 + 2] * invN;
  const float* cx = coords + (size_t)(b * 3 + 0) * N;
  const float* cy = coords + (size_t)(b * 3 + 1) * N;
  const float* cz = coords + (size_t)(b * 3 + 2) * N;
  float m = 0.f;
  for (int i = blockIdx.x * blockDim.x + tid; i < N; i += gridDim.x * blockDim.x) {
    float dx = cx[i] - mx, dy = cy[i] - my, dz = cz[i] - mz;
    m = fmaxf(m, sqrtf(dx * dx + dy * dy + dz * dz));
  }
  __shared__ float red[256];
  red[tid] = m;
  __syncthreads();
  for (int off = 128; off > 0; off >>= 1) {
    if (tid < off) red[tid] = fmaxf(red[tid], red[tid + off]);
    __syncthreads();
  }
  if (tid == 0) {
    unsigned* mn = (unsigned*)ws + WS_MAXN + b;
    atomicMax(mn, __float_as_uint(red[0]));  // nonneg floats: uint order == float order
  }
}

// ---------- kernel 3: per-point normalize, emit norm_coords, voxel index, counts ----------
__global__ void k_points(const float* __restrict__ coords, float* __restrict__ ws,
                         float* __restrict__ out, int N) {
  const int idx = blockIdx.x * blockDim.x + threadIdx.x;
  if (idx >= BATCH * N) return;
  const int b = idx / N;
  const int n = idx - b * N;
  const float invN = 1.0f / (float)N;
  const float mx = ws[WS_SUMS + b * 4 + 0] * invN;
  const float my = ws[WS_SUMS + b * 4 + 1] * invN;
  const float mz = ws[WS_SUMS + b * 4 + 2] * invN;
  const float maxn = __uint_as_float(((const unsigned*)ws)[WS_MAXN + b]);
  const float inv_sc = 1.0f / (maxn * 2.0f);   // EPS == 0

  float c0 = coords[(size_t)(b * 3 + 0) * N + n];
  float c1 = coords[(size_t)(b * 3 + 1) * N + n];
  float c2 = coords[(size_t)(b * 3 + 2) * N + n];
  float n0 = fminf(fmaxf(((c0 - mx) * inv_sc + 0.5f) * (float)RES, 0.f), (float)(RES - 1));
  float n1 = fminf(fmaxf(((c1 - my) * inv_sc + 0.5f) * (float)RES, 0.f), (float)(RES - 1));
  float n2 = fminf(fmaxf(((c2 - mz) * inv_sc + 0.5f) * (float)RES, 0.f), (float)(RES - 1));

  float* nc = out + GRID_ELEMS;                // norm_coords region, (B,3,N)
  nc[(size_t)(b * 3 + 0) * N + n] = n0;
  nc[(size_t)(b * 3 + 1) * N + n] = n1;
  nc[(size_t)(b * 3 + 2) * N + n] = n2;

  int vx = (int)rintf(n0), vy = (int)rintf(n1), vz = (int)rintf(n2);  // round-to-nearest-even
  int vrel = vx * (RES * RES) + vy * RES + vz;
  ((int*)ws)[WS_VREL + idx] = vrel;
  atomicAdd(&ws[WS_COUNTS + b * R3 + vrel], 1.0f);
}

// ---------- kernel 4: LDS-privatized segment sum per (b,c), normalize, async writeback ----------
__global__ void k_grid(const float* __restrict__ features, const float* __restrict__ ws,
                       float* __restrict__ out, int N) {
  extern __shared__ float lds[];               // R3 floats = 128 KB (fits 320 KB WGP LDS)
  const int b = blockIdx.y;
  const int c = blockIdx.x;
  const int tid = threadIdx.x;
  const int nthr = blockDim.x;

  for (int i = tid; i < R3; i += nthr) lds[i] = 0.0f;
  __syncthreads();

  const float* feat = features + (size_t)(b * CH + c) * N;
  const int* vr = (const int*)ws + WS_VREL + (size_t)b * N;   // L2-resident (3.2 MB total)
  for (int i = tid; i < N; i += nthr) {
    if (i + 4096 < N) __builtin_prefetch(&feat[i + 4096], 0, 0);  // global_prefetch
    int v = vr[i];
    float f = feat[i];
    atomicAdd(&lds[v], f);                     // ds_add_f32, no global atomics
  }
  __syncthreads();

  // normalize slice in LDS
  const float* cnt = ws + WS_COUNTS + (size_t)b * R3;
  for (int i = tid * 4; i < R3; i += nthr * 4) {
    float4 s = *(float4*)&lds[i];
    const float4 k = *(const float4*)&cnt[i];
    s.x /= fmaxf(k.x, 1.0f);
    s.y /= fmaxf(k.y, 1.0f);
    s.z /= fmaxf(k.z, 1.0f);
    s.w /= fmaxf(k.w, 1.0f);
    *(float4*)&lds[i] = s;
  }
  __syncthreads();

  // async LDS -> global writeback of the (b,c) slice in final (B,C,x,y,z) layout
  float* dst = out + (size_t)(b * CH + c) * R3;
  for (int i = tid * 4; i < R3; i += nthr * 4) {
    async_store_lds_to_global_b128(&dst[i], &lds[i]);
  }
  wait_asynccnt0();
}

extern "C" void kernel_launch(void* const* d_in, const int* in_sizes, int n_in,
                              void* d_out, int out_size, void* d_ws, size_t ws_size,
                              hipStream_t stream) {
  const float* features = (const float*)d_in[0];   // (B, C, N) f32
  const float* coords   = (const float*)d_in[1];   // (B, 3, N) f32
  float* out = (float*)d_out;                      // grid (B,C,32,32,32) ++ norm_coords (B,3,N)
  float* ws  = (float*)d_ws;
  const int N = in_sizes[1] / (3 * BATCH);

  k_init<<<(WS_INIT_WORDS + 255) / 256, 256, 0, stream>>>(ws);

  dim3 gr(64, BATCH);
  k_sums<<<gr, 256, 0, stream>>>(coords, ws, N);
  k_maxnorm<<<gr, 256, 0, stream>>>(coords, ws, N);

  const int total = BATCH * N;
  k_points<<<(total + 255) / 256, 256, 0, stream>>>(coords, ws, out, N);

  dim3 ge(CH, BATCH);  // 512 workgroups, each privatizing one 128 KB voxel slice
  k_grid<<<ge, 512, R3 * sizeof(float), stream>>>(features, ws, out, N);
}